// dynamics_model_4294967296291
// MI455X (gfx1250) — compile-verified
//
#include <hip/hip_runtime.h>
#include <math.h>

// ---------------------------------------------------------------------------
// Lyapunov-stable dynamics model forward (CDNA5 / gfx1250, wave32)
//   fhat: 32 -> 25 -> 25 -> 32 softplus MLP (scalar VALU, cheap)
//   V(z) = ||g(z)-g(0)||^2 + eps*||z||^2, g: 32 ->128 ->128 -> 8
//   Newton+bisection rootfind for gamma on violating samples.
// One wave handles 16 samples; V-MLP layers run on V_WMMA_F32_16X16X4_F32
// with all weights resident in LDS in B-fragment layout.
// veval is noinline (2 instances) to avoid scratch spills; LDS is reached
// via extern __shared__ + integer offsets so all traffic stays ds_*.
// ---------------------------------------------------------------------------

typedef __attribute__((ext_vector_type(2))) float v2f;
typedef __attribute__((ext_vector_type(8))) float v8f;

#define BETA_C   0.99f
#define TOL_C    1e-4f
#define MAXIT_C  1000
#define EPS_C    1e-3f
#define NDIM     32
#define HFD      25
#define WAVES_PER_BLOCK 4

// ---- shared memory layout (in floats) ----
constexpr int OFF_W1P  = 0;                    // 2048 v2f  = 4096 f   (W1 k-pair packed)
constexpr int OFF_W2P  = 4096;                 // 8192 v2f  = 16384 f  (W2 k-pair packed)
constexpr int OFF_W3T  = OFF_W2P + 16384;      // 1024 f    (W3 transposed [o][k])
constexpr int OFF_B1   = OFF_W3T + 1024;       // 128
constexpr int OFF_B2   = OFF_B1 + 128;         // 128
constexpr int OFF_B3   = OFF_B2 + 128;         // 8
constexpr int OFF_H0   = OFF_B3 + 8;           // 8   (g(0))
constexpr int OFF_FW1  = OFF_H0 + 8;           // 800
constexpr int OFF_FB1  = OFF_FW1 + 800;        // 32 (25 used)
constexpr int OFF_FW2  = OFF_FB1 + 32;         // 640 (625 used)
constexpr int OFF_FB2  = OFF_FW2 + 640;        // 32 (25 used)
constexpr int OFF_FW3  = OFF_FB2 + 32;         // 800
constexpr int OFF_FB3  = OFF_FW3 + 800;        // 32
constexpr int OFF_HTMP = OFF_FB3 + 32;         // 256 (h0 precompute temp)
constexpr int OFF_WAVE = OFF_HTMP + 256;       // per-wave scratch from here
constexpr int WAVE_FLOATS = 9216;              // xbuf512 fbuf512 h2048 t2048 h2b2048 t2b2048
constexpr int SMEM_FLOATS = OFF_WAVE + WAVES_PER_BLOCK * WAVE_FLOATS;

__device__ __forceinline__ float softplus_f(float a) {
  return a > 20.0f ? a : log1pf(expf(a));
}

struct VR { float V; float dV; };

// Evaluate V(s * in) (and dV/ds if DV) for the wave's 16 samples.
// `inOff`  : float offset (into dynamic LDS) of the 16x32 input tile
// `waveOff`: float offset of this wave's scratch region
// `s`      : per-lane scale (gamma of sample lane&15; replicated lanes m,m+16)
// `f2`     : ||in||^2 of sample lane&15 (replicated)
// Result replicated in lanes m and m+16. Must be called with EXEC all-ones.
template<bool DV>
__device__ __attribute__((noinline)) VR veval(int waveOff, int inOff, float s, float f2) {
  extern __shared__ float smem[];
  const v2f*   w1p = (const v2f*)(smem + OFF_W1P);
  const v2f*   w2p = (const v2f*)(smem + OFF_W2P);
  const float* w3t = smem + OFF_W3T;
  const float* b1s = smem + OFF_B1;
  const float* b2s = smem + OFF_B2;
  const float* b3s = smem + OFF_B3;
  const float* h0s = smem + OFF_H0;
  const float* inb = smem + inOff;
  float* hbuf  = smem + waveOff + 1024;
  float* tbuf  = smem + waveOff + 3072;
  float* h2buf = smem + waveOff + 5120;
  float* t2buf = smem + waveOff + 7168;

  const int lane = threadIdx.x & 31;
  const int m    = lane & 15;
  const int half = lane >> 4;

  // ---- layer 1: (16x32) @ (32x128) ----
#pragma unroll 1
  for (int nt = 0; nt < 8; ++nt) {
    v8f C  = {};
    v8f Ct = {};
#pragma unroll
    for (int kc = 0; kc < 8; ++kc) {
      v2f a = *(const v2f*)(inb + m * 32 + kc * 4 + half * 2);
      v2f as; as[0] = a[0] * s; as[1] = a[1] * s;
      v2f b = w1p[(2 * kc + half) * 128 + nt * 16 + m];
      C = __builtin_amdgcn_wmma_f32_16x16x4_f32(false, as, false, b, (short)0, C, false, false);
      if (DV)
        Ct = __builtin_amdgcn_wmma_f32_16x16x4_f32(false, a, false, b, (short)0, Ct, false, false);
    }
    float bias = b1s[nt * 16 + m];
#pragma unroll
    for (int j = 0; j < 8; ++j) {
      float hv = softplus_f(C[j] + bias);
      int idx = (j + 8 * half) * 128 + nt * 16 + m;   // C layout: row j+8*half, col nt*16+m
      hbuf[idx] = hv;
      if (DV) tbuf[idx] = (1.0f - expf(-hv)) * Ct[j];  // sigmoid(a)=1-e^{-softplus(a)}
    }
  }
  // wave-private LDS buffers; DS ops are in-order within a wave -> no barrier.

  // ---- layer 2: (16x128) @ (128x128) ----
#pragma unroll 1
  for (int nt = 0; nt < 8; ++nt) {
    v8f C  = {};
    v8f Ct = {};
#pragma unroll 8
    for (int kc = 0; kc < 32; ++kc) {
      v2f a = *(const v2f*)(hbuf + m * 128 + kc * 4 + half * 2);
      v2f b = w2p[(2 * kc + half) * 128 + nt * 16 + m];
      C = __builtin_amdgcn_wmma_f32_16x16x4_f32(false, a, false, b, (short)0, C, false, false);
      if (DV) {
        v2f at = *(const v2f*)(tbuf + m * 128 + kc * 4 + half * 2);
        Ct = __builtin_amdgcn_wmma_f32_16x16x4_f32(false, at, false, b, (short)0, Ct, false, false);
      }
    }
    float bias = b2s[nt * 16 + m];
#pragma unroll
    for (int j = 0; j < 8; ++j) {
      float hv = softplus_f(C[j] + bias);
      int idx = (j + 8 * half) * 128 + nt * 16 + m;
      h2buf[idx] = hv;
      if (DV) t2buf[idx] = (1.0f - expf(-hv)) * Ct[j];
    }
  }

  // ---- layer 3: (16x128) @ (128x8), scalar from LDS ----
  // lane handles sample m, outputs o = half*4 .. half*4+3; halves combined via shfl_xor(16)
  float Vp = 0.0f, dVp = 0.0f;
#pragma unroll 1
  for (int oi = 0; oi < 4; ++oi) {
    int o = half * 4 + oi;
    float acc = b3s[o];
    float tac = 0.0f;
#pragma unroll 8
    for (int k = 0; k < 128; k += 4) {
      float4 hv = *(const float4*)(h2buf + m * 128 + k);
      float4 wv = *(const float4*)(w3t + o * 128 + k);
      acc += hv.x * wv.x + hv.y * wv.y + hv.z * wv.z + hv.w * wv.w;
      if (DV) {
        float4 tv = *(const float4*)(t2buf + m * 128 + k);
        tac += tv.x * wv.x + tv.y * wv.y + tv.z * wv.z + tv.w * wv.w;
      }
    }
    float d = acc - h0s[o];
    Vp += d * d;
    if (DV) dVp += d * tac;
  }
  Vp += __shfl_xor(Vp, 16);
  VR r;
  r.V = Vp + EPS_C * s * s * f2;
  r.dV = 0.0f;
  if (DV) {
    dVp += __shfl_xor(dVp, 16);
    r.dV = 2.0f * dVp + 2.0f * EPS_C * s * f2;
  }
  return r;
}

extern "C" __global__ void lyap_forward_kernel(
    const float* __restrict__ x,
    const float* __restrict__ fW1, const float* __restrict__ fb1,
    const float* __restrict__ fW2, const float* __restrict__ fb2,
    const float* __restrict__ fW3, const float* __restrict__ fb3,
    const float* __restrict__ vW1, const float* __restrict__ vb1,
    const float* __restrict__ vW2, const float* __restrict__ vb2,
    const float* __restrict__ vW3, const float* __restrict__ vb3,
    float* __restrict__ out) {
  extern __shared__ float smem[];
  v2f*   w1p  = (v2f*)(smem + OFF_W1P);
  v2f*   w2p  = (v2f*)(smem + OFF_W2P);
  float* w3t  = smem + OFF_W3T;
  float* b1s  = smem + OFF_B1;
  float* b2s  = smem + OFF_B2;
  float* b3s  = smem + OFF_B3;
  float* h0s  = smem + OFF_H0;
  float* fw1s = smem + OFF_FW1;
  float* fb1s = smem + OFF_FB1;
  float* fw2s = smem + OFF_FW2;
  float* fb2s = smem + OFF_FB2;
  float* fw3s = smem + OFF_FW3;
  float* fb3s = smem + OFF_FB3;
  float* htmp = smem + OFF_HTMP;

  const int tid = threadIdx.x;

  // ---- cooperative weight staging (W packed to B-fragment k-pair layout) ----
  for (int i = tid; i < 2048; i += blockDim.x) {     // W1: 32x128 -> 16 k-pairs
    int p = i >> 7, n = i & 127;
    v2f t; t[0] = vW1[(2 * p) * 128 + n]; t[1] = vW1[(2 * p + 1) * 128 + n];
    w1p[i] = t;
  }
  for (int i = tid; i < 8192; i += blockDim.x) {     // W2: 128x128 -> 64 k-pairs
    int p = i >> 7, n = i & 127;
    v2f t; t[0] = vW2[(2 * p) * 128 + n]; t[1] = vW2[(2 * p + 1) * 128 + n];
    w2p[i] = t;
  }
  for (int i = tid; i < 1024; i += blockDim.x) {     // W3 transposed [o][k]
    int o = i >> 7, k = i & 127;
    w3t[i] = vW3[k * 8 + o];
  }
  for (int i = tid; i < 128; i += blockDim.x) { b1s[i] = vb1[i]; b2s[i] = vb2[i]; }
  if (tid < 8) b3s[tid] = vb3[tid];
  for (int i = tid; i < 800; i += blockDim.x) fw1s[i] = fW1[i];
  for (int i = tid; i < 625; i += blockDim.x) fw2s[i] = fW2[i];
  for (int i = tid; i < 800; i += blockDim.x) fw3s[i] = fW3[i];
  for (int i = tid; i < HFD; i += blockDim.x) { fb1s[i] = fb1[i]; fb2s[i] = fb2[i]; }
  for (int i = tid; i < NDIM; i += blockDim.x) fb3s[i] = fb3[i];
  __syncthreads();

  // ---- h0 = g(0) head, recomputed per block (trivial cost) ----
  float* h1t = htmp;
  float* h2t = htmp + 128;
  if (tid < 128) h1t[tid] = softplus_f(b1s[tid]);
  __syncthreads();
  if (tid < 128) {
    float a = b2s[tid];
#pragma unroll 8
    for (int k = 0; k < 128; ++k) a += h1t[k] * w2p[(k >> 1) * 128 + tid][k & 1];
    h2t[tid] = softplus_f(a);
  }
  __syncthreads();
  if (tid < 8) {
    float a = b3s[tid];
#pragma unroll 8
    for (int k = 0; k < 128; ++k) a += h2t[k] * w3t[tid * 128 + k];
    h0s[tid] = a;
  }
  __syncthreads();

  // ---- per-wave tile of 16 samples ----
  const int lane = tid & 31;
  const int wid  = tid >> 5;
  const int m    = lane & 15;
  const int half = lane >> 4;
  const int base = (blockIdx.x * WAVES_PER_BLOCK + wid) * 16;

  const int waveOff = OFF_WAVE + wid * WAVE_FLOATS;
  const int xOff    = waveOff;        // 16x32 x tile
  const int fOff    = waveOff + 512;  // 16x32 fhatx tile
  float* xbuf = smem + xOff;
  float* fbuf = smem + fOff;

  for (int i = lane; i < 512; i += 32) xbuf[i] = x[base * 32 + i];

  // ---- fhat per sample (lanes 0..15 each own one sample) ----
  if (lane < 16) {
    const float* xr = xbuf + lane * 32;
    float h1[HFD], h2[HFD];
#pragma unroll 1
    for (int j = 0; j < HFD; ++j) {
      float a = fb1s[j];
#pragma unroll 8
      for (int k = 0; k < NDIM; ++k) a += xr[k] * fw1s[k * HFD + j];
      h1[j] = softplus_f(a);
    }
#pragma unroll 1
    for (int j = 0; j < HFD; ++j) {
      float a = fb2s[j];
#pragma unroll 5
      for (int k = 0; k < HFD; ++k) a += h1[k] * fw2s[k * HFD + j];
      h2[j] = softplus_f(a);
    }
#pragma unroll 1
    for (int n = 0; n < NDIM; ++n) {
      float a = fb3s[n];
#pragma unroll 5
      for (int j = 0; j < HFD; ++j) a += h2[j] * fw3s[j * NDIM + n];
      fbuf[lane * 32 + n] = a;
    }
  }

  // ---- squared norms per sample (replicated lanes m, m+16) ----
  float x2 = 0.0f, f2 = 0.0f;
#pragma unroll
  for (int kc = 0; kc < 8; ++kc) {
    int k = kc * 4 + half * 2;
    v2f xv = *(const v2f*)(xbuf + m * 32 + k);
    v2f fv = *(const v2f*)(fbuf + m * 32 + k);
    x2 += xv[0] * xv[0] + xv[1] * xv[1];
    f2 += fv[0] * fv[0] + fv[1] * fv[1];
  }
  x2 += __shfl_xor(x2, 16);
  f2 += __shfl_xor(f2, 16);

  // ---- target and violation check ----
  float Vx = veval<false>(waveOff, xOff, 1.0f, x2).V;
  float target = BETA_C * Vx;
  float Vf = veval<false>(waveOff, fOff, 1.0f, f2).V;

  bool viol   = Vf > target;
  bool active = viol;
  float gamma = 1.0f, e1 = 0.0f, e2 = 1.0f;

  // ---- masked Newton + bisection rootfind (wave-uniform control flow) ----
  int it = 0;
  while (__ballot((int)active) != 0ull && it < MAXIT_C) {
    VR r = veval<true>(waveOff, fOff, gamma, f2);
    float newt = gamma - (r.V - target) / r.dV;
    float g2 = active ? newt : gamma;
    bool bisec = active && (g2 < e1 || g2 > e2 || !__builtin_isfinite(g2));
    if (__ballot((int)bisec) != 0ull) {
      float a = 0.5f * (e1 + e2);
      float sv[3]; sv[0] = a; sv[1] = e1; sv[2] = e2;
      float Vv[3];
#pragma unroll 1
      for (int q = 0; q < 3; ++q) Vv[q] = veval<false>(waveOff, fOff, sv[q], f2).V;
      float sa = (Vv[0] > target) ? 1.0f : ((Vv[0] < target) ? -1.0f : 0.0f);
      float s1 = (Vv[1] > target) ? 1.0f : ((Vv[1] < target) ? -1.0f : 0.0f);
      float s2 = (Vv[2] > target) ? 1.0f : ((Vv[2] < target) ? -1.0f : 0.0f);
      if (bisec && sa * s1 < 0.0f) e2 = a;
      if (bisec && sa * s2 < 0.0f) e1 = a;
      g2 = bisec ? a : g2;
    }
    gamma = g2;
    float Vc = veval<false>(waveOff, fOff, gamma, f2).V;
    active = active && (fabsf(Vc - target) > TOL_C);
    ++it;
  }

  // ---- output: fhatx * (viol ? gamma : 1) ----
  float oscale = viol ? gamma : 1.0f;   // lane j (j<16) holds sample j's scale
  for (int i = lane; i < 512; i += 32) {
    float sc = __shfl(oscale, i >> 5);
    out[base * 32 + i] = fbuf[i] * sc;
  }
}

extern "C" void kernel_launch(void* const* d_in, const int* in_sizes, int n_in,
                              void* d_out, int out_size, void* d_ws, size_t ws_size,
                              hipStream_t stream) {
  (void)n_in; (void)out_size; (void)d_ws; (void)ws_size;
  const float* x   = (const float*)d_in[0];
  const float* fW1 = (const float*)d_in[1];
  const float* fb1 = (const float*)d_in[2];
  const float* fW2 = (const float*)d_in[3];
  const float* fb2 = (const float*)d_in[4];
  const float* fW3 = (const float*)d_in[5];
  const float* fb3 = (const float*)d_in[6];
  const float* vW1 = (const float*)d_in[7];
  const float* vb1 = (const float*)d_in[8];
  const float* vW2 = (const float*)d_in[9];
  const float* vb2 = (const float*)d_in[10];
  const float* vW3 = (const float*)d_in[11];
  const float* vb3 = (const float*)d_in[12];
  float* out = (float*)d_out;

  const int B = in_sizes[0] / NDIM;               // 65536
  const int blocks = B / (16 * WAVES_PER_BLOCK);  // 1024
  const size_t shmem = (size_t)SMEM_FLOATS * sizeof(float);  // ~239 KB < 320 KB WGP LDS

  hipLaunchKernelGGL(lyap_forward_kernel, dim3(blocks), dim3(WAVES_PER_BLOCK * 32),
                     shmem, stream,
                     x, fW1, fb1, fW2, fb2, fW3, fb3,
                     vW1, vb1, vW2, vb2, vW3, vb3, out);
}